// TimeSpatialTransformer_59803124629642
// MI455X (gfx1250) — compile-verified
//
#include <hip/hip_runtime.h>
#include <math.h>

#define B_   16
#define W_   2048
#define C_   512
#define NTOT ((size_t)B_ * W_ * C_)
#define MAXJ 64

typedef float v8f __attribute__((ext_vector_type(8)));
typedef float v2f __attribute__((ext_vector_type(2)));

__device__ __forceinline__ float softplusf(float x) {
  // stable: max(x,0) + log1p(exp(-|x|))  (== jax.nn.softplus)
  return fmaxf(x, 0.0f) + log1pf(expf(-fabsf(x)));
}

// ---------------- K1: inverse in-degree per (b, column j) ----------------
__global__ __launch_bounds__(256)
void k_indeg(const float* __restrict__ adj, float* __restrict__ inv_deg) {
  int idx = blockIdx.x * 256 + threadIdx.x;        // b*C + j
  int b = idx >> 9;                                // / C_
  int j = idx & (C_ - 1);
  const float* p = adj + (size_t)b * C_ * C_ + j;
  float s = 0.0f;
  #pragma unroll 4
  for (int i = 0; i < C_; ++i) s += fmaxf(p[(size_t)i * C_], 0.0f);
  inv_deg[idx] = 1.0f / fmaxf(s, 1e-4f);
}

// ---------------- K2: X_spatial = X_sfr @ (relu(adj) * inv_deg[col]) ----
// Block tile 128(M) x 128(N), K chunks of 32; 8 waves as 4(M) x 2(N).
// Each wave: 2 M-subtiles x 4 N-subtiles (32x64) of V_WMMA_F32_16X16X4_F32.
// A staged row-major As[m][k] (stride 36); B staged TRANSPOSED Bst[n][k]
// (stride 34) so each lane's {K,K+1} pair is one ds_load_b64.
// f32 A 16x4 layout: lanes 0-15 M=lane {K0,K1}, lanes 16-31 M=lane-16 {K2,K3}.
// f32 B 4x16 layout: lanes 0-15 N=lane {K0,K1}, lanes 16-31 N=lane-16 {K2,K3}.
// f32 C/D 16x16: VGPR r -> row r (lanes 0-15) / row r+8 (lanes 16-31).
constexpr int BK  = 32;
constexpr int LDA = 36;
constexpr int LDBT = 34;

__device__ __forceinline__ void load_tiles(const float* __restrict__ Ag,
                                           const float* __restrict__ Bg,
                                           int k0, int tid, int m0, int n0,
                                           float4* Areg, float* Breg) {
  #pragma unroll
  for (int it = 0; it < 4; ++it) {                 // A: 128x32 = 1024 float4
    int idx = tid + it * 256;
    int r = idx >> 3;
    int kk = (idx & 7) << 2;
    Areg[it] = *(const float4*)(Ag + (size_t)(m0 + r) * C_ + (k0 + kk));
  }
  const int col = tid & 127;                       // B: 32x128 scalar, coalesced
  const int rb  = tid >> 7;
  #pragma unroll
  for (int it = 0; it < 16; ++it) {
    int r = it * 2 + rb;
    Breg[it] = fmaxf(Bg[(size_t)(k0 + r) * C_ + (n0 + col)], 0.0f);
  }
}

__global__ __launch_bounds__(256)
void k_spatial_gemm(const float* __restrict__ X, const float* __restrict__ adj,
                    const float* __restrict__ inv_deg, float* __restrict__ Xs) {
  __shared__ float As[128 * LDA];
  __shared__ float Bst[128 * LDBT];

  const int b  = blockIdx.z;
  const int m0 = blockIdx.y * 128;
  const int n0 = blockIdx.x * 128;
  const float* Ag = X   + (size_t)b * W_ * C_;
  const float* Bg = adj + (size_t)b * C_ * C_;

  const int tid   = threadIdx.x;
  const int wave  = tid >> 5;
  const int waveM = wave >> 1;      // 0..3
  const int waveN = wave & 1;       // 0..1
  const int lane  = tid & 31;
  const int lhalf = lane >> 4;      // 0 or 1
  const int l16   = lane & 15;

  v8f acc[2][4] = {};
  float4 Areg[4];
  float  Breg[16];

  load_tiles(Ag, Bg, 0, tid, m0, n0, Areg, Breg);

  const int rA0  = waveM * 32 + l16;
  const int rA1  = rA0 + 16;
  const int koff = lhalf << 1;
  const int colb = waveN * 64 + l16;

  for (int k0 = 0; k0 < C_; k0 += BK) {
    __syncthreads();                 // previous chunk's compute finished
    // stage registers -> LDS
    #pragma unroll
    for (int it = 0; it < 4; ++it) {
      int idx = tid + it * 256;
      int r = idx >> 3;
      int kk = (idx & 7) << 2;
      float* d = &As[r * LDA + kk];
      float4 v = Areg[it];
      d[0] = v.x; d[1] = v.y; d[2] = v.z; d[3] = v.w;
    }
    {
      const int col = tid & 127;
      const int rb  = tid >> 7;
      #pragma unroll
      for (int it = 0; it < 16; ++it)
        Bst[col * LDBT + it * 2 + rb] = Breg[it];
    }
    __syncthreads();
    // prefetch next chunk while WMMAs run
    if (k0 + BK < C_)
      load_tiles(Ag, Bg, k0 + BK, tid, m0, n0, Areg, Breg);

    #pragma unroll
    for (int k = 0; k < BK; k += 4) {
      v2f a0, a1;
      a0.x = As[rA0 * LDA + k + koff];
      a0.y = As[rA0 * LDA + k + koff + 1];
      a1.x = As[rA1 * LDA + k + koff];
      a1.y = As[rA1 * LDA + k + koff + 1];
      #pragma unroll
      for (int nt = 0; nt < 4; ++nt) {
        v2f bb;
        bb.x = Bst[(colb + nt * 16) * LDBT + k + koff];
        bb.y = Bst[(colb + nt * 16) * LDBT + k + koff + 1];
        acc[0][nt] = __builtin_amdgcn_wmma_f32_16x16x4_f32(
            false, a0, false, bb, (short)0, acc[0][nt], false, false);
        acc[1][nt] = __builtin_amdgcn_wmma_f32_16x16x4_f32(
            false, a1, false, bb, (short)0, acc[1][nt], false, false);
      }
    }
  }

  // epilogue: column scale + store
  #pragma unroll
  for (int s = 0; s < 2; ++s) {
    const int rowb = m0 + waveM * 32 + s * 16 + lhalf * 8;
    #pragma unroll
    for (int nt = 0; nt < 4; ++nt) {
      int ncol = n0 + waveN * 64 + nt * 16 + l16;
      float sc = inv_deg[b * C_ + ncol];
      float* outp = Xs + (size_t)b * W_ * C_ + ncol;
      #pragma unroll
      for (int r = 0; r < 8; ++r)
        outp[(size_t)(rowb + r) * C_] = acc[s][nt][r] * sc;
    }
  }
}

// ---------------- K3: temporal token = mean over channels ----------------
__global__ __launch_bounds__(256)
void k_token(const float* __restrict__ Xs, float* __restrict__ tt) {
  int row = blockIdx.x;                 // b*W + i
  const float* p = Xs + (size_t)row * C_;
  float s = p[threadIdx.x] + p[threadIdx.x + 256];
  #pragma unroll
  for (int o = 16; o > 0; o >>= 1) s += __shfl_down(s, o, 32);
  __shared__ float part[8];
  if ((threadIdx.x & 31) == 0) part[threadIdx.x >> 5] = s;
  __syncthreads();
  if (threadIdx.x == 0) {
    float t = 0.0f;
    #pragma unroll
    for (int w = 0; w < 8; ++w) t += part[w];
    tt[row] = t * (1.0f / C_);
  }
}

// ---------------- K4: banded rank-1 attention + fuse + outputs ----------
__global__ __launch_bounds__(256)
void k_attn_fuse(const float* __restrict__ Xs, const float* __restrict__ tt,
                 const float* __restrict__ log_temp,
                 const float* __restrict__ wq, const float* __restrict__ bq,
                 const float* __restrict__ wk, const float* __restrict__ bk,
                 const float* __restrict__ wv, const float* __restrict__ bv,
                 const float* __restrict__ wmu, const float* __restrict__ bmu,
                 const float* __restrict__ wsig, const float* __restrict__ bsig,
                 const int* __restrict__ k1p, const int* __restrict__ k2p,
                 float* __restrict__ out) {
  __shared__ float wgt[MAXJ];
  __shared__ int   jl[MAXJ];
  __shared__ int   cnt_s;
  __shared__ float sumw_s;

  const int row = blockIdx.x;           // b*W + i
  const int b = row >> 11;              // / W_
  const int i = row & (W_ - 1);

  if (threadIdx.x == 0) {
    const int k1 = *k1p, k2 = *k2p;
    int cnt = 0;
    // j in [i-k2, i-k1]
    for (int j = i - k2; j <= i - k1 && cnt < MAXJ; ++j)
      if (j >= 0 && j < W_) jl[cnt++] = j;
    // j in [i+k1, i+k2] (skip j==i dup when k1==0: first loop already has it)
    int lo2 = i + ((k1 > 0) ? k1 : 1);
    for (int j = lo2; j <= i + k2 && cnt < MAXJ; ++j)
      if (j >= 0 && j < W_) jl[cnt++] = j;
    if (cnt == 0) { jl[0] = i; cnt = 1; }   // identity-row fallback

    const float temp = softplusf(log_temp[0]) + 1e-4f;
    const float qv = wq[0] * tt[(size_t)b * W_ + i] + bq[0];
    float mx = -3.4e38f;
    for (int t = 0; t < cnt; ++t) {
      float kv = wk[0] * tt[(size_t)b * W_ + jl[t]] + bk[0];
      float lv = qv * kv / temp;
      wgt[t] = lv;
      mx = fmaxf(mx, lv);
    }
    float S = 0.0f;
    for (int t = 0; t < cnt; ++t) { float e = expf(wgt[t] - mx); wgt[t] = e; S += e; }
    float S1 = 0.0f;
    for (int t = 0; t < cnt; ++t) { float p = wgt[t] / S; wgt[t] = p; S1 += p; }
    float denom = S1 + 1e-6f;
    for (int t = 0; t < cnt; ++t) wgt[t] = wgt[t] / denom;
    sumw_s = S1 / denom;                 // == attn.sum(-1) == sacon scalar
    cnt_s = cnt;
  }
  __syncthreads();

  const int cnt = cnt_s;
  const float sumw = sumw_s;
  const float wvv = wv[0],  bvv = bv[0];
  const float wm  = wmu[0], bm  = bmu[0];
  const float wsg = wsig[0], bsg = bsig[0];
  const float* base = Xs + (size_t)b * W_ * C_;

  #pragma unroll
  for (int cc = 0; cc < 2; ++cc) {
    int c = threadIdx.x + cc * 256;
    float dot = 0.0f;
    for (int t = 0; t < cnt; ++t) dot += wgt[t] * base[(size_t)jl[t] * C_ + c];
    // X_temporal = sum_t w_t * (wv*Xs[j]+bv) = wv*dot + bv*sum(w)
    float xt = wvv * dot + bvv * sumw;
    float fused = xt + base[(size_t)i * C_ + c];
    float mu = fminf(fmaxf(wm * fused + bm, -20.0f), 20.0f);
    float sg = softplusf(wsg * fused + bsg) + 0.1f;
    size_t idx = (size_t)row * C_ + c;
    out[idx]             = mu;
    out[NTOT + idx]      = sg;
    out[2 * NTOT + idx]  = sumw;
  }
}

// ---------------- host ----------------
extern "C" void kernel_launch(void* const* d_in, const int* in_sizes, int n_in,
                              void* d_out, int out_size, void* d_ws, size_t ws_size,
                              hipStream_t stream) {
  (void)in_sizes; (void)n_in; (void)out_size; (void)ws_size;
  const float* X        = (const float*)d_in[0];
  const float* adj      = (const float*)d_in[1];
  const float* log_temp = (const float*)d_in[2];
  const float* wq   = (const float*)d_in[3];
  const float* bq   = (const float*)d_in[4];
  const float* wk   = (const float*)d_in[5];
  const float* bk   = (const float*)d_in[6];
  const float* wv   = (const float*)d_in[7];
  const float* bv   = (const float*)d_in[8];
  const float* wmu  = (const float*)d_in[9];
  const float* bmu  = (const float*)d_in[10];
  const float* wsig = (const float*)d_in[11];
  const float* bsig = (const float*)d_in[12];
  const int*   k1   = (const int*)d_in[13];
  const int*   k2   = (const int*)d_in[14];
  float* out = (float*)d_out;

  // workspace layout: X_spatial [B,W,C] | tt [B,W] | inv_deg [B,C]
  float* Xs      = (float*)d_ws;
  float* tt      = Xs + NTOT;
  float* inv_deg = tt + (size_t)B_ * W_;

  k_indeg<<<(B_ * C_) / 256, 256, 0, stream>>>(adj, inv_deg);
  k_spatial_gemm<<<dim3(C_ / 128, W_ / 128, B_), 256, 0, stream>>>(X, adj, inv_deg, Xs);
  k_token<<<B_ * W_, 256, 0, stream>>>(Xs, tt);
  k_attn_fuse<<<B_ * W_, 256, 0, stream>>>(Xs, tt, log_temp, wq, bq, wk, bk,
                                           wv, bv, wmu, bmu, wsig, bsig,
                                           k1, k2, out);
}